// HeteroGNN_GAT_For_Embeddings_3066606649344
// MI455X (gfx1250) — compile-verified
//
#include <hip/hip_runtime.h>
#include <math.h>

#define HID 128
#define HEADS 4
#define CH 32

typedef float v2f __attribute__((ext_vector_type(2)));
typedef float v8f __attribute__((ext_vector_type(8)));

// ---------------------------------------------------------------------------
// CDNA5 async global->LDS copy (ASYNCcnt-tracked), per CDNA5 ISA
// Each lane copies 16 bytes from the global address g to the LDS offset lds_off.
// ---------------------------------------------------------------------------
__device__ __forceinline__ void async_g2l_b128(unsigned lds_off, const float* g) {
  asm volatile("global_load_async_to_lds_b128 %0, %1, off"
               :
               : "v"(lds_off), "v"(g)
               : "memory");
}
__device__ __forceinline__ void wait_async0() {
  asm volatile("s_wait_asynccnt 0" ::: "memory");
}
// low 32 bits of a flat shared-aperture address == LDS offset (ISA 10.2)
__device__ __forceinline__ unsigned lds_off32(const void* p) {
  return (unsigned)(uintptr_t)p;
}

// ---------------------------------------------------------------------------
// float atomic max via int/uint atomics (works with -inf init)
// ---------------------------------------------------------------------------
__device__ __forceinline__ void atomic_max_f32(float* addr, float val) {
  if (val >= 0.0f)
    atomicMax((int*)addr, __float_as_int(val));
  else
    atomicMin((unsigned int*)addr, __float_as_uint(val));
}

// ---------------------------------------------------------------------------
// f32 WMMA GEMM: C[M,N] = A[M,K] @ B[K,N] (+bias, optional relu)
// One block = one 16-row M-tile; wave w computes N-tile w. A tile (16xK,
// pad 4) and full B slab (KxN, pad 16) are staged to LDS with
// global_load_async_to_lds_b128; fragments come from LDS (conflict-free).
// V_WMMA_F32_16X16X4_F32, K fully unrolled.
//   A-frag: lane<16 -> M=lane, K={k0,k0+1}; lane>=16 -> M=lane-16, K={k0+2,k0+3}
//   B-frag: mirrored (per-lane-half K contiguous across the 2 VGPRs)
//   C/D:    vgpr j -> M = 16*mt + 8*(lane>=16) + j, N = 16*nt + (lane&15)
// ---------------------------------------------------------------------------
template <int K, int N>
__global__ __launch_bounds__((N / 16) * 32) void wmma_gemm_f32(
    const float* __restrict__ A, const float* __restrict__ B,
    const float* __restrict__ bias, float* __restrict__ C, int M, int relu) {
  constexpr int THREADS = (N / 16) * 32;
  constexpr int KP = K + 4;   // A row pad: ds_load_b64 frags conflict-free
  constexpr int NP = N + 16;  // B row pad: lane-halves hit disjoint bank sets
  __shared__ __attribute__((aligned(16))) float As[16 * KP];
  __shared__ __attribute__((aligned(16))) float Bs[K * NP];

  const int tid = threadIdx.x;
  const int lane = tid & 31;
  const int nt = tid >> 5;       // N-tile of this wave
  const int mt = blockIdx.x;     // M-tile of this block
  const int hi = lane >> 4;
  const int lo = lane & 15;

  // ---- stage A tile: 16 rows x K floats (4K b128 vectors), coalesced ----
  {
    constexpr int VECS = 4 * K;  // (16*K)/4
    for (int i = tid; i < VECS; i += THREADS) {
      const int r = i / (K / 4);
      const int cv = i - r * (K / 4);
      int row = mt * 16 + r;
      if (row > M - 1) row = M - 1;  // ragged tail: duplicate last row
      async_g2l_b128(lds_off32(&As[r * KP + cv * 4]),
                     A + (size_t)row * K + cv * 4);
    }
  }
  // ---- stage B slab: K rows x N floats, coalesced ----
  {
    constexpr int VECS = (K * N) / 4;
    for (int i = tid; i < VECS; i += THREADS) {
      const int r = i / (N / 4);
      const int cv = i - r * (N / 4);
      async_g2l_b128(lds_off32(&Bs[r * NP + cv * 4]),
                     B + (size_t)r * N + cv * 4);
    }
  }
  wait_async0();
  __syncthreads();

  const float* __restrict__ Al = As + lo * KP;
  const float* __restrict__ Bl = Bs + nt * 16 + lo;

  v8f acc = {};
#pragma unroll
  for (int k0 = 0; k0 < K; k0 += 4) {
    const int ka = k0 + 2 * hi;
    v2f a;
    a.x = Al[ka];
    a.y = Al[ka + 1];
    v2f b;
    b.x = Bl[ka * NP];
    b.y = Bl[(ka + 1) * NP];
    acc = __builtin_amdgcn_wmma_f32_16x16x4_f32(false, a, false, b, (short)0,
                                                acc, false, false);
  }

  const int col = nt * 16 + lo;
  const float bv = bias ? bias[col] : 0.0f;
#pragma unroll
  for (int j = 0; j < 8; ++j) {
    const int row = mt * 16 + hi * 8 + j;
    if (row < M) {
      float v = acc[j] + bv;
      if (relu) v = fmaxf(v, 0.0f);
      C[(size_t)row * N + col] = v;
    }
  }
}

// ---------------------------------------------------------------------------
// V[k,h] = sum_c W[k, h*32+c] * a[h*32+c]   (folds attention vec thru weight)
// ---------------------------------------------------------------------------
__global__ void head_vec(const float* __restrict__ W, const float* __restrict__ a,
                         float* __restrict__ V) {
  int t = blockIdx.x * blockDim.x + threadIdx.x;
  if (t >= HID * HEADS) return;
  int k = t >> 2, h = t & 3;
  float s = 0.0f;
  const float* wr = W + k * HID + h * CH;
  const float* ar = a + h * CH;
#pragma unroll
  for (int c = 0; c < CH; ++c) s += wr[c] * ar[c];
  V[k * HEADS + h] = s;
}

// S[n,h] = x[n,:] . V[:,h]
__global__ void node_scores(const float* __restrict__ X, const float* __restrict__ V,
                            float* __restrict__ S, int Nn) {
  int n = blockIdx.x * blockDim.x + threadIdx.x;
  if (n >= Nn) return;
  float s0 = 0, s1 = 0, s2 = 0, s3 = 0;
  const float* x = X + (size_t)n * HID;
#pragma unroll 4
  for (int k = 0; k < HID; ++k) {
    float xv = x[k];
    const float* vr = V + k * HEADS;
    s0 += xv * vr[0];
    s1 += xv * vr[1];
    s2 += xv * vr[2];
    s3 += xv * vr[3];
  }
  float* sp = S + (size_t)n * HEADS;
  sp[0] = s0; sp[1] = s1; sp[2] = s2; sp[3] = s3;
}

// ---------------------------------------------------------------------------
// small utility kernels
// ---------------------------------------------------------------------------
__global__ void fill_val(float* __restrict__ p, float v, size_t n) {
  size_t t = (size_t)blockIdx.x * blockDim.x + threadIdx.x;
  if (t < n) p[t] = v;
}

__global__ void init_bias_rows(float* __restrict__ out, const float* __restrict__ b0,
                               const float* __restrict__ b1, size_t total) {
  size_t t = (size_t)blockIdx.x * blockDim.x + threadIdx.x;
  if (t >= total) return;
  int j = (int)(t & (HID - 1));
  float v = b0[j];
  if (b1) v += b1[j];
  out[t] = v;
}

__global__ void relu_inplace(float* __restrict__ p, size_t n) {
  size_t t = (size_t)blockIdx.x * blockDim.x + threadIdx.x;
  if (t < n) p[t] = fmaxf(p[t], 0.0f);
}

__global__ void gather_rows(const float* __restrict__ Emb, const int* __restrict__ idx,
                            float* __restrict__ O, int Nn, int rows) {
  size_t t = (size_t)blockIdx.x * blockDim.x + threadIdx.x;
  if (t >= (size_t)Nn * HID) return;
  int n = (int)(t >> 7), j = (int)(t & (HID - 1));
  int i = idx[n];
  i = i < 0 ? 0 : (i > rows - 1 ? rows - 1 : i);
  O[t] = Emb[(size_t)i * HID + j];
}

// ---------------------------------------------------------------------------
// edge passes for segment softmax + aggregation
// ---------------------------------------------------------------------------
__global__ void edge_logits(const float* __restrict__ Ssrc, const float* __restrict__ Sdst,
                            const int* __restrict__ src, const int* __restrict__ dst,
                            float* __restrict__ Ebuf, float* __restrict__ Mbuf, int ne) {
  int e = blockIdx.x * blockDim.x + threadIdx.x;
  if (e >= ne) return;
  int s = src[e], d = dst[e];
#pragma unroll
  for (int h = 0; h < HEADS; ++h) {
    float v = Ssrc[(size_t)s * HEADS + h] + Sdst[(size_t)d * HEADS + h];
    v = v > 0.0f ? v : 0.2f * v;  // leaky_relu(0.2)
    Ebuf[(size_t)e * HEADS + h] = v;
    atomic_max_f32(&Mbuf[(size_t)d * HEADS + h], v);
  }
}

__global__ void edge_softmax_num(const int* __restrict__ dst, float* __restrict__ Ebuf,
                                 const float* __restrict__ Mbuf, float* __restrict__ Den,
                                 int ne) {
  int e = blockIdx.x * blockDim.x + threadIdx.x;
  if (e >= ne) return;
  int d = dst[e];
#pragma unroll
  for (int h = 0; h < HEADS; ++h) {
    float m = Mbuf[(size_t)d * HEADS + h];
    if (!(m > -3.0e38f)) m = 0.0f;  // where(isfinite(m), m, 0)
    float p = expf(Ebuf[(size_t)e * HEADS + h] - m);
    Ebuf[(size_t)e * HEADS + h] = p;
    atomicAdd(&Den[(size_t)d * HEADS + h], p);
  }
}

// 128 threads = 1 channel each; block strides over `epb` edges -> coalesced hs
__global__ __launch_bounds__(128) void edge_aggregate(
    const int* __restrict__ src, const int* __restrict__ dst,
    const float* __restrict__ P, const float* __restrict__ Den,
    const float* __restrict__ HS, float* __restrict__ Out, int ne, int epb) {
  const int j = threadIdx.x;
  const int h = j >> 5;
  int e0 = blockIdx.x * epb;
  int e1 = e0 + epb;
  if (e1 > ne) e1 = ne;
  for (int e = e0; e < e1; ++e) {
    int s = src[e], d = dst[e];
    if (e + 1 < e1) {  // hide next gather's latency (global_prefetch_b8)
      __builtin_prefetch(&HS[(size_t)src[e + 1] * HID + j], 0, 0);
    }
    float coef = P[(size_t)e * HEADS + h] / (Den[(size_t)d * HEADS + h] + 1e-16f);
    atomicAdd(&Out[(size_t)d * HID + j], coef * HS[(size_t)s * HID + j]);
  }
}

// ---------------------------------------------------------------------------
// host orchestration
// ---------------------------------------------------------------------------
static inline int cdiv(long long a, long long b) { return (int)((a + b - 1) / b); }

static void run_gemm(const float* A, const float* B, const float* bias, float* C,
                     int M, int N, int K, int relu, hipStream_t s) {
  int mtiles = cdiv(M, 16);
  if (K == 32 && N == 128)
    wmma_gemm_f32<32, 128><<<mtiles, 256, 0, s>>>(A, B, bias, C, M, relu);
  else if (K == 128 && N == 128)
    wmma_gemm_f32<128, 128><<<mtiles, 256, 0, s>>>(A, B, bias, C, M, relu);
  else if (K == 128 && N == 64)
    wmma_gemm_f32<128, 64><<<mtiles, 128, 0, s>>>(A, B, bias, C, M, relu);
}

struct Ctx {
  float *hs, *ssrc, *sdst, *ebuf, *mbuf, *den, *vs, *vd;
  int ne;
};

static void run_gat(const float* xs, int Ns, const float* xd, int Nd,
                    const int* src, const int* dst, int n_dst,
                    const float* Ws, const float* Wd, const float* As, const float* Ad,
                    float* out_accum, const Ctx& c, hipStream_t s) {
  head_vec<<<2, 256, 0, s>>>(Ws, As, c.vs);
  head_vec<<<2, 256, 0, s>>>(Wd, Ad, c.vd);
  run_gemm(xs, Ws, nullptr, c.hs, Ns, HID, HID, 0, s);  // hs = xs @ Ws
  node_scores<<<cdiv(Ns, 256), 256, 0, s>>>(xs, c.vs, c.ssrc, Ns);
  node_scores<<<cdiv(Nd, 256), 256, 0, s>>>(xd, c.vd, c.sdst, Nd);
  size_t nd4 = (size_t)n_dst * HEADS;
  fill_val<<<cdiv(nd4, 256), 256, 0, s>>>(c.mbuf, -__builtin_inff(), nd4);
  fill_val<<<cdiv(nd4, 256), 256, 0, s>>>(c.den, 0.0f, nd4);
  edge_logits<<<cdiv(c.ne, 256), 256, 0, s>>>(c.ssrc, c.sdst, src, dst, c.ebuf, c.mbuf, c.ne);
  edge_softmax_num<<<cdiv(c.ne, 256), 256, 0, s>>>(dst, c.ebuf, c.mbuf, c.den, c.ne);
  const int epb = 8;
  edge_aggregate<<<cdiv(c.ne, epb), 128, 0, s>>>(src, dst, c.ebuf, c.den, c.hs,
                                                 out_accum, c.ne, epb);
}

extern "C" void kernel_launch(void* const* d_in, const int* in_sizes, int n_in,
                              void* d_out, int out_size, void* d_ws, size_t ws_size,
                              hipStream_t stream) {
  const int* user_idx = (const int*)d_in[0];
  const int* merch_idx = (const int*)d_in[1];
  const float* tx_feat = (const float*)d_in[2];
  const int* eut_src = (const int*)d_in[3];
  const int* eut_dst = (const int*)d_in[4];
  const int* etm_src = (const int*)d_in[5];
  const int* etm_dst = (const int*)d_in[6];
  const float* user_emb = (const float*)d_in[7];
  const float* merch_emb = (const float*)d_in[8];
  const float* tx_W = (const float*)d_in[9];
  const float* tx_b = (const float*)d_in[10];
  const float* Wsrc = (const float*)d_in[11];
  const float* Wdst = (const float*)d_in[12];
  const float* Asrc = (const float*)d_in[13];
  const float* Adst = (const float*)d_in[14];
  const float* Bias = (const float*)d_in[15];
  const float* proj_W = (const float*)d_in[16];
  const float* proj_b = (const float*)d_in[17];
  float* out = (float*)d_out;

  const int n_user = in_sizes[0];
  const int n_merch = in_sizes[1];
  const int n_tx = in_sizes[2] / 32;
  const int ne = in_sizes[3];

  // workspace carve-up (floats)
  float* ws = (float*)d_ws;
  size_t o = 0;
  float* hu = ws + o;  o += (size_t)n_user * HID;
  float* hm = ws + o;  o += (size_t)n_merch * HID;
  float* ht = ws + o;  o += (size_t)n_tx * HID;
  float* hu2 = ws + o; o += (size_t)n_user * HID;
  float* hm2 = ws + o; o += (size_t)n_merch * HID;
  float* ht2 = ws + o; o += (size_t)n_tx * HID;
  Ctx c;
  c.hs = ws + o;   o += (size_t)n_tx * HID;     // max Ns
  c.ssrc = ws + o; o += (size_t)n_tx * HEADS;
  c.sdst = ws + o; o += (size_t)n_tx * HEADS;
  c.ebuf = ws + o; o += (size_t)ne * HEADS;
  c.mbuf = ws + o; o += (size_t)n_tx * HEADS;   // max n_dst
  c.den = ws + o;  o += (size_t)n_tx * HEADS;
  c.vs = ws + o;   o += HID * HEADS;
  c.vd = ws + o;   o += HID * HEADS;
  c.ne = ne;

  // initial features
  gather_rows<<<cdiv((size_t)n_user * HID, 256), 256, 0, stream>>>(user_emb, user_idx, hu, n_user, n_user);
  gather_rows<<<cdiv((size_t)n_merch * HID, 256), 256, 0, stream>>>(merch_emb, merch_idx, hm, n_merch, n_merch);
  run_gemm(tx_feat, tx_W, tx_b, ht, n_tx, HID, 32, /*relu=*/1, stream);

  const size_t WQ = (size_t)HID * HID;   // per (l,r) weight block
  const size_t AQ = (size_t)HEADS * CH;  // per (l,r) attention block

  for (int l = 0; l < 2; ++l) {
    const int q = l * 4;
    init_bias_rows<<<cdiv((size_t)n_tx * HID, 256), 256, 0, stream>>>(
        ht2, Bias + (size_t)(q + 0) * HID, Bias + (size_t)(q + 1) * HID, (size_t)n_tx * HID);
    init_bias_rows<<<cdiv((size_t)n_user * HID, 256), 256, 0, stream>>>(
        hu2, Bias + (size_t)(q + 2) * HID, nullptr, (size_t)n_user * HID);
    init_bias_rows<<<cdiv((size_t)n_merch * HID, 256), 256, 0, stream>>>(
        hm2, Bias + (size_t)(q + 3) * HID, nullptr, (size_t)n_merch * HID);

    // r0: users -> tx
    run_gat(hu, n_user, ht, n_tx, eut_src, eut_dst, n_tx,
            Wsrc + (q + 0) * WQ, Wdst + (q + 0) * WQ, Asrc + (q + 0) * AQ, Adst + (q + 0) * AQ,
            ht2, c, stream);
    // r1: merchants -> tx  (src=etm_dst, dst=etm_src per reference)
    run_gat(hm, n_merch, ht, n_tx, etm_dst, etm_src, n_tx,
            Wsrc + (q + 1) * WQ, Wdst + (q + 1) * WQ, Asrc + (q + 1) * AQ, Adst + (q + 1) * AQ,
            ht2, c, stream);
    // r2: tx -> users
    run_gat(ht, n_tx, hu, n_user, eut_dst, eut_src, n_user,
            Wsrc + (q + 2) * WQ, Wdst + (q + 2) * WQ, Asrc + (q + 2) * AQ, Adst + (q + 2) * AQ,
            hu2, c, stream);
    // r3: tx -> merchants
    run_gat(ht, n_tx, hm, n_merch, etm_src, etm_dst, n_merch,
            Wsrc + (q + 3) * WQ, Wdst + (q + 3) * WQ, Asrc + (q + 3) * AQ, Adst + (q + 3) * AQ,
            hm2, c, stream);

    relu_inplace<<<cdiv((size_t)n_tx * HID, 256), 256, 0, stream>>>(ht2, (size_t)n_tx * HID);
    relu_inplace<<<cdiv((size_t)n_user * HID, 256), 256, 0, stream>>>(hu2, (size_t)n_user * HID);
    relu_inplace<<<cdiv((size_t)n_merch * HID, 256), 256, 0, stream>>>(hm2, (size_t)n_merch * HID);

    float* t;
    t = hu; hu = hu2; hu2 = t;
    t = hm; hm = hm2; hm2 = t;
    t = ht; ht = ht2; ht2 = t;
  }

  // final projection: out = ht @ proj_W + proj_b  -> [n_tx, 64]
  run_gemm(ht, proj_W, proj_b, out, n_tx, 64, HID, /*relu=*/0, stream);
}